// GraphDeformationBlock_17609365914513
// MI455X (gfx1250) — compile-verified
//
#include <hip/hip_runtime.h>

typedef __attribute__((ext_vector_type(2))) float v2f;
typedef __attribute__((ext_vector_type(8))) float v8f;

#define BS 256

__global__ void fill_f32(float* __restrict__ p, int n, float v) {
    int i = blockIdx.x * blockDim.x + threadIdx.x;
    if (i < n) p[i] = v;
}

__global__ void deg_accum(const int* __restrict__ dst, float* __restrict__ deg, int ne) {
    int i = blockIdx.x * blockDim.x + threadIdx.x;
    if (i < ne) atomicAdd(&deg[dst[i]], 1.0f);
}

__global__ void inv_deg_k(float* __restrict__ deg, int nv) {
    int i = blockIdx.x * blockDim.x + threadIdx.x;
    if (i < nv) deg[i] = 1.0f / fmaxf(deg[i], 1.0f);
}

// Zero-pad a KxN row-major matrix into KpxNp.
__global__ void pack_pad(const float* __restrict__ src, float* __restrict__ dst,
                         int K, int N, int Kp, int Np) {
    int i = blockIdx.x * blockDim.x + threadIdx.x;
    int total = Kp * Np;
    if (i >= total) return;
    int k = i / Np, n = i - k * Np;
    dst[i] = (k < K && n < N) ? src[k * N + n] : 0.0f;
}

// Trilinear sample of (16,128,128,128) grid + concat [verts/128, feats] -> X0 stride 20 (col 19 = 0 pad)
__global__ void trilinear_concat(const float* __restrict__ img, const float* __restrict__ verts,
                                 float* __restrict__ X0, int nv) {
    int i = blockIdx.x * blockDim.x + threadIdx.x;
    if (i >= nv) return;
    const int G = 128, G2 = G * G, G3 = G2 * G;
    float vx = verts[i * 3 + 0], vy = verts[i * 3 + 1], vz = verts[i * 3 + 2];
    float px = fminf(fmaxf(vx, 0.0f), 127.0f);
    float py = fminf(fmaxf(vy, 0.0f), 127.0f);
    float pz = fminf(fmaxf(vz, 0.0f), 127.0f);
    float fx = floorf(px), fy = floorf(py), fz = floorf(pz);
    int x0 = (int)fx, y0 = (int)fy, z0 = (int)fz;
    int x1 = min(x0 + 1, 127), y1 = min(y0 + 1, 127), z1 = min(z0 + 1, 127);
    float wx = px - fx, wy = py - fy, wz = pz - fz;
    float* xr = X0 + (size_t)i * 20;
    xr[0] = vx * (1.0f / 128.0f);
    xr[1] = vy * (1.0f / 128.0f);
    xr[2] = vz * (1.0f / 128.0f);
    int b000 = x0 * G2 + y0 * G + z0;
    int b001 = x0 * G2 + y0 * G + z1;
    int b010 = x0 * G2 + y1 * G + z0;
    int b011 = x0 * G2 + y1 * G + z1;
    int b100 = x1 * G2 + y0 * G + z0;
    int b101 = x1 * G2 + y0 * G + z1;
    int b110 = x1 * G2 + y1 * G + z0;
    int b111 = x1 * G2 + y1 * G + z1;
#pragma unroll 4
    for (int c = 0; c < 16; ++c) {
        const float* base = img + (size_t)c * G3;
        float c00 = base[b000] * (1.0f - wz) + base[b001] * wz;
        float c01 = base[b010] * (1.0f - wz) + base[b011] * wz;
        float c10 = base[b100] * (1.0f - wz) + base[b101] * wz;
        float c11 = base[b110] * (1.0f - wz) + base[b111] * wz;
        xr[3 + c] = (c00 * (1.0f - wy) + c01 * wy) * (1.0f - wx) +
                    (c10 * (1.0f - wy) + c11 * wy) * wx;
    }
    xr[19] = 0.0f;
}

// nb[dst] += x[src], thread per (edge, 4-feature group)
__global__ void scatter_edges(const float* __restrict__ X, const int* __restrict__ src,
                              const int* __restrict__ dst, float* __restrict__ NB,
                              int ne, int Kp, int ng) {
    int tid = blockIdx.x * blockDim.x + threadIdx.x;
    int e = tid / ng;
    if (e >= ne) return;
    int g = tid - e * ng;
    int s = src[e], d = dst[e];
    const float4 v = *(const float4*)(X + (size_t)s * Kp + g * 4);
    float* o = NB + (size_t)d * Kp + g * 4;
    atomicAdd(o + 0, v.x);
    atomicAdd(o + 1, v.y);
    atomicAdd(o + 2, v.z);
    atomicAdd(o + 3, v.w);
}

// One wave per 16x16 output tile: D = X@Ws + (NB*invdeg)@Wn, then bias (+leaky-relu or final write)
__global__ void edgeconv_gemm(const float* __restrict__ X, const float* __restrict__ NB,
                              const float* __restrict__ Ws, const float* __restrict__ Wn,
                              const float* __restrict__ Bp, const float* __restrict__ invdeg,
                              float* __restrict__ Y,
                              const float* __restrict__ verts, float* __restrict__ out,
                              int K, int N, int nv, int finalLayer) {
    int gtid = blockIdx.x * blockDim.x + threadIdx.x;
    int wave = gtid >> 5;                 // wave-uniform
    int lane = threadIdx.x & 31;
    int colTiles = N >> 4;
    int rowTiles = (nv + 15) >> 4;
    if (wave >= rowTiles * colTiles) return;   // uniform guard -> EXEC all-ones below
    int rowTile = wave / colTiles;
    int colTile = wave - rowTile * colTiles;
    int row0 = rowTile << 4;
    int col0 = colTile << 4;
    int r = lane & 15;
    int half = lane >> 4;
    int arow = row0 + r;
    if (arow >= nv) arow = nv - 1;        // keep EXEC full; dup rows are store-guarded
    float idg = invdeg[arow];
    const float* xr = X + (size_t)arow * K;
    const float* nr = NB + (size_t)arow * K;
    v8f c = {0.f, 0.f, 0.f, 0.f, 0.f, 0.f, 0.f, 0.f};
    for (int k0 = 0; k0 < K; k0 += 4) {
        int kk = k0 + half * 2;           // A 16x4 layout: lanes 0-15 K={k0,k0+1}, lanes 16-31 K={k0+2,k0+3}
        v2f a_s; a_s.x = xr[kk]; a_s.y = xr[kk + 1];
        v2f b_s; b_s.x = Ws[kk * N + col0 + r]; b_s.y = Ws[(kk + 1) * N + col0 + r];
        c = __builtin_amdgcn_wmma_f32_16x16x4_f32(false, a_s, false, b_s, (short)0, c, false, false);
        v2f a_n; a_n.x = nr[kk] * idg; a_n.y = nr[kk + 1] * idg;
        v2f b_n; b_n.x = Wn[kk * N + col0 + r]; b_n.y = Wn[(kk + 1) * N + col0 + r];
        c = __builtin_amdgcn_wmma_f32_16x16x4_f32(false, a_n, false, b_n, (short)0, c, false, false);
    }
    int col = col0 + r;
    float bias = Bp[col];
#pragma unroll
    for (int vv = 0; vv < 8; ++vv) {
        int row = row0 + vv + half * 8;   // D layout: lanes 0-15 M=vgpr, lanes 16-31 M=vgpr+8
        if (row >= nv) continue;
        float val = c[vv] + bias;
        if (finalLayer) {
            if (col < 3) out[row * 3 + col] = verts[row * 3 + col] + 0.1f * val;
        } else {
            val = val > 0.0f ? val : 0.3f * val;  // leaky_relu slope 0.3
            Y[(size_t)row * N + col] = val;
        }
    }
}

static inline int nblk(long n, int bs) { return (int)((n + bs - 1) / bs); }

extern "C" void kernel_launch(void* const* d_in, const int* in_sizes, int n_in,
                              void* d_out, int out_size, void* d_ws, size_t ws_size,
                              hipStream_t stream) {
    const float* img   = (const float*)d_in[0];
    const float* verts = (const float*)d_in[1];
    const int*   esrc  = (const int*)d_in[2];
    const int*   edst  = (const int*)d_in[3];
    const float *wself[4], *wnb[4], *bias[4];
    for (int i = 0; i < 4; ++i) {
        wself[i] = (const float*)d_in[4 + 3 * i];
        wnb[i]   = (const float*)d_in[5 + 3 * i];
        bias[i]  = (const float*)d_in[6 + 3 * i];
    }
    int nv = in_sizes[1] / 3;
    int ne = in_sizes[2];

    // workspace layout (floats)
    float* ws     = (float*)d_ws;
    float* invdeg = ws;                        // nv
    float* Xb     = invdeg + nv;               // nv*64
    float* Yb     = Xb + (size_t)nv * 64;      // nv*64
    float* NBb    = Yb + (size_t)nv * 64;      // nv*64
    float* wbuf   = NBb + (size_t)nv * 64;     // packed weights + biases

    const int Kp[4] = {20, 32, 64, 64};        // padded K (A stride)
    const int Np[4] = {32, 64, 64, 16};        // padded N
    const int K0[4] = {19, 32, 64, 64};
    const int N0[4] = {32, 64, 64, 3};
    float* pw[4][2]; float* pb[4];
    {
        float* p = wbuf;
        for (int i = 0; i < 4; ++i) {
            pw[i][0] = p; p += Kp[i] * Np[i];
            pw[i][1] = p; p += Kp[i] * Np[i];
            pb[i]    = p; p += Np[i];
        }
    }

    // 1) degrees -> inv_deg
    fill_f32<<<nblk(nv, BS), BS, 0, stream>>>(invdeg, nv, 0.0f);
    deg_accum<<<nblk(ne, BS), BS, 0, stream>>>(edst, invdeg, ne);
    inv_deg_k<<<nblk(nv, BS), BS, 0, stream>>>(invdeg, nv);

    // 2) pack weights/biases to padded shapes
    for (int i = 0; i < 4; ++i) {
        int tot = Kp[i] * Np[i];
        pack_pad<<<nblk(tot, BS), BS, 0, stream>>>(wself[i], pw[i][0], K0[i], N0[i], Kp[i], Np[i]);
        pack_pad<<<nblk(tot, BS), BS, 0, stream>>>(wnb[i],   pw[i][1], K0[i], N0[i], Kp[i], Np[i]);
        pack_pad<<<nblk(Np[i], BS), BS, 0, stream>>>(bias[i], pb[i], 1, N0[i], 1, Np[i]);
    }

    // 3) trilinear sample + concat -> Xb (stride 20)
    trilinear_concat<<<nblk(nv, BS), BS, 0, stream>>>(img, verts, Xb, nv);

    // 4) four edge-conv layers (ping-pong activations)
    const float* layerIn[4] = {Xb, Yb, Xb, Yb};
    float*       layerOut[4] = {Yb, Xb, Yb, nullptr};
    for (int i = 0; i < 4; ++i) {
        int K = Kp[i], N = Np[i], ng = K / 4;
        fill_f32<<<nblk((long)nv * K, BS), BS, 0, stream>>>(NBb, nv * K, 0.0f);
        scatter_edges<<<nblk((long)ne * ng, BS), BS, 0, stream>>>(layerIn[i], esrc, edst, NBb, ne, K, ng);
        long waves = (long)((nv + 15) / 16) * (N / 16);
        int gblocks = (int)((waves * 32 + BS - 1) / BS);
        edgeconv_gemm<<<gblocks, BS, 0, stream>>>(layerIn[i], NBb, pw[i][0], pw[i][1], pb[i], invdeg,
                                                  layerOut[i], verts, (float*)d_out,
                                                  K, N, nv, (i == 3) ? 1 : 0);
    }
}